// Aaren_9208409882812
// MI455X (gfx1250) — compile-verified
//
#include <hip/hip_runtime.h>
#include <math.h>

// Aaren online-softmax scan, L=65536, D=256, fp32.
// Memory-bound (~192MB min traffic -> ~8us @ 23.3TB/s). Reduce-then-scan:
//   k_gemv: s = K@q            (streams K once)
//   k_agg : per-group (m,u,w)  (streams V once)
//   k_scan: exclusive scan of 512 group aggregates (1 block, channel-parallel)
//   k_out : tile replay; 16x16 triangular softmax-weight matrix A times V-tile
//           done with V_WMMA_F32_16X16X4_F32 chains (full fp32), V re-read hits L2.

#define LSEQ 65536
#define DM   256
#define NG   512          // groups
#define RPG  128          // rows per group (LSEQ/NG)
#define NTILE 8           // 16-row tiles per group (RPG/16)

typedef __attribute__((ext_vector_type(2))) float v2f;
typedef __attribute__((ext_vector_type(8))) float v8f;

// ---------------- Kernel 1: s = K @ q  (one wave per row) ----------------
__global__ __launch_bounds__(256) void k_gemv(const float* __restrict__ K,
                                              const float* __restrict__ q,
                                              float* __restrict__ s) {
  __shared__ float qs[DM];
  int t = threadIdx.x;
  qs[t] = q[t];
  __syncthreads();
  int wv = t >> 5, lane = t & 31;
  int row = blockIdx.x * 8 + wv;
  const float4* k4 = reinterpret_cast<const float4*>(K + (size_t)row * DM + lane * 8);
  const float4* q4 = reinterpret_cast<const float4*>(&qs[lane * 8]);
  float4 ka = k4[0], kb = k4[1];
  float4 qa = q4[0], qb = q4[1];
  float p = ka.x*qa.x + ka.y*qa.y + ka.z*qa.z + ka.w*qa.w
          + kb.x*qb.x + kb.y*qb.y + kb.z*qb.z + kb.w*qb.w;
  #pragma unroll
  for (int off = 16; off >= 1; off >>= 1) p += __shfl_xor(p, off, 32);
  if (lane == 0) s[row] = p;
}

// ---------------- Kernel 2: per-group aggregate (m, u, w[256]) ----------------
__global__ __launch_bounds__(256) void k_agg(const float* __restrict__ s,
                                             const float* __restrict__ V,
                                             float* __restrict__ agg_m,
                                             float* __restrict__ agg_u,
                                             float* __restrict__ agg_w) {
  __shared__ float sb[RPG];
  int t = threadIdx.x, g = blockIdx.x;
  if (t < RPG) sb[t] = s[g * RPG + t];
  __syncthreads();
  float m = -INFINITY;
  #pragma unroll 8
  for (int i = 0; i < RPG; ++i) m = fmaxf(m, sb[i]);
  float u = 0.f, w = 0.f;
  const float* vp = V + (size_t)g * RPG * DM + t;   // thread t = channel d
  #pragma unroll 4
  for (int i = 0; i < RPG; ++i) {
    float e = __expf(sb[i] - m);
    u += e;
    w += vp[(size_t)i * DM] * e;
  }
  agg_w[g * DM + t] = w;
  if (t == 0) { agg_m[g] = m; agg_u[g] = u; }
}

// ---------------- Kernel 3: exclusive scan over group aggregates ----------------
__global__ __launch_bounds__(256) void k_scan(const float* __restrict__ agg_m,
                                              const float* __restrict__ agg_u,
                                              const float* __restrict__ agg_w,
                                              float* __restrict__ pre_m,
                                              float* __restrict__ pre_u,
                                              float* __restrict__ pre_w) {
  int t = threadIdx.x;                 // channel d
  float M = -INFINITY, U = 0.f, W = 0.f;
  for (int b = 0; b < NG; ++b) {
    pre_w[b * DM + t] = W;
    if (t == 0) { pre_m[b] = M; pre_u[b] = U; }
    float mb = agg_m[b], ub = agg_u[b], wb = agg_w[b * DM + t];
    float Mn = fmaxf(M, mb);
    float eP = __expf(M - Mn), eC = __expf(mb - Mn);
    U = U * eP + ub * eC;
    W = W * eP + wb * eC;
    M = Mn;
  }
}

// ---------------- Kernel 4: tile replay + WMMA triangular matmul + output ----------------
__global__ __launch_bounds__(256) void k_out(const float* __restrict__ s,
                                             const float* __restrict__ V,
                                             const float* __restrict__ pre_m,
                                             const float* __restrict__ pre_u,
                                             const float* __restrict__ pre_w,
                                             float* __restrict__ out) {
  __shared__ float Wst[DM];          // carried numerator state W[d]
  __shared__ float Al[16 * 17];      // A[i][j], padded row stride 17 (no bank conflicts)
  __shared__ float stile[16];
  __shared__ float marr[16], iuarr[16], parr[16];  // running max, 1/u, prefix scale
  __shared__ float Ms, Us;

  int t = threadIdx.x, g = blockIdx.x;
  int lane = t & 31, wv = t >> 5;
  int hi = (lane >= 16) ? 1 : 0;
  int l15 = lane & 15;
  int n0 = wv * 32;                  // each wave owns column tiles [n0,n0+16) and [n0+16,n0+32)
  int n1 = n0 + 16;

  Wst[t] = pre_w[g * DM + t];
  if (t == 0) { Ms = pre_m[g]; Us = pre_u[g]; }
  __syncthreads();

  for (int tt = 0; tt < NTILE; ++tt) {
    int r0 = g * RPG + tt * 16;

    // Prefetch this tile's V operands (independent of scan state -> overlaps
    // the serial recurrence below). B layout: lane<16 holds K-rows {4k,4k+1},
    // lane>=16 holds {4k+2,4k+3}, N = lane&15.
    float b0x[4], b0y[4], b1x[4], b1y[4];
    #pragma unroll
    for (int kk = 0; kk < 4; ++kk) {
      int krow = kk * 4 + hi * 2;
      const float* vb = V + (size_t)(r0 + krow) * DM;
      b0x[kk] = vb[n0 + l15];
      b0y[kk] = vb[DM + n0 + l15];
      b1x[kk] = vb[n1 + l15];
      b1y[kk] = vb[DM + n1 + l15];
    }

    if (t < 16) stile[t] = s[r0 + t];
    __syncthreads();

    // Serial 16-step online-softmax recurrence (scalars only), thread 0.
    // 16 IEEE divides happen HERE (once per row), not per-thread in the epilogue.
    if (t == 0) {
      float m = Ms, u = Us, m0 = Ms;
      #pragma unroll
      for (int i = 0; i < 16; ++i) {
        float si = stile[i];
        float mn = fmaxf(m, si);
        u = u * __expf(m - mn) + __expf(si - mn);
        m = mn;
        marr[i]  = mn;               // running global max
        iuarr[i] = 1.0f / u;         // reciprocal denominator
        parr[i]  = __expf(m0 - mn);  // prefix-W scale (<= 1)
      }
      Ms = m; Us = u;
    }
    __syncthreads();

    // Build triangular weight matrix A[i][j] = (j<=i) ? exp(s_j - m_i) : 0
    {
      int i = t >> 4, j = t & 15;
      Al[i * 17 + j] = (j <= i) ? __expf(stile[j] - marr[i]) : 0.f;
    }
    __syncthreads();

    // (A x V_tile) via 4 chained f32 WMMAs per 16-wide column tile.
    v8f acc0 = {}; v8f acc1 = {};
    #pragma unroll
    for (int kk = 0; kk < 4; ++kk) {
      int acol = kk * 4 + hi * 2;
      v2f a; a.x = Al[l15 * 17 + acol]; a.y = Al[l15 * 17 + acol + 1];
      v2f b0; b0.x = b0x[kk]; b0.y = b0y[kk];
      v2f b1; b1.x = b1x[kk]; b1.y = b1y[kk];
      acc0 = __builtin_amdgcn_wmma_f32_16x16x4_f32(false, a, false, b0, (short)0, acc0, false, false);
      acc1 = __builtin_amdgcn_wmma_f32_16x16x4_f32(false, a, false, b1, (short)0, acc1, false, false);
    }

    // out[i][d] = (P_i * W[d] + (A V)[i][d]) * (1/u_i)
    // C/D layout: VGPR r -> row r + 8*hi, col = lane&15.
    float w0 = Wst[n0 + l15];        // loop-invariant across the 8 rows
    float w1 = Wst[n1 + l15];
    #pragma unroll
    for (int r = 0; r < 8; ++r) {
      int row = r + hi * 8;
      float pw = parr[row], iu = iuarr[row];
      size_t o = (size_t)(r0 + row) * DM;
      out[o + n0 + l15] = (w0 * pw + acc0[r]) * iu;
      out[o + n1 + l15] = (w1 * pw + acc1[r]) * iu;
    }
    __syncthreads();

    // Carry state: W' = P_15 * W + (A V)[15]; row 15 lives in acc[7] of hi lanes.
    if (hi) {
      float p15 = parr[15];
      Wst[n0 + l15] = w0 * p15 + acc0[7];
      Wst[n1 + l15] = w1 * p15 + acc1[7];
    }
    __syncthreads();
  }
}

// ---------------- Host launcher ----------------
extern "C" void kernel_launch(void* const* d_in, const int* in_sizes, int n_in,
                              void* d_out, int out_size, void* d_ws, size_t ws_size,
                              hipStream_t stream) {
  (void)in_sizes; (void)n_in; (void)out_size; (void)ws_size;
  const float* K = (const float*)d_in[0];
  const float* V = (const float*)d_in[1];
  const float* q = (const float*)d_in[2];
  float* out = (float*)d_out;

  float* ws = (float*)d_ws;
  float* s   = ws;                       // LSEQ
  float* am  = s  + LSEQ;                // NG
  float* au  = am + NG;                  // NG
  float* aw  = au + NG;                  // NG*DM
  float* pm  = aw + (size_t)NG * DM;     // NG
  float* pu  = pm + NG;                  // NG
  float* pw  = pu + NG;                  // NG*DM
  // total ws: 65536 + 2*(2*512 + 512*256) floats ~= 1.3 MB

  k_gemv<<<LSEQ / 8, 256, 0, stream>>>(K, q, s);
  k_agg <<<NG,       256, 0, stream>>>(s, V, am, au, aw);
  k_scan<<<1,        256, 0, stream>>>(am, au, aw, pm, pu, pw);
  k_out <<<NG,       256, 0, stream>>>(s, V, pm, pu, pw, out);
}